// HOGLayer_29901562315052
// MI455X (gfx1250) — compile-verified
//
#include <hip/hip_runtime.h>
#include <math.h>

#define NBINS  10
#define TILE   64            // interior pixels per block (64x64)
#define TROWS  66            // tile + 1-halo each side
#define TPITCH 68            // padded LDS row pitch (floats); 272B = 17*16B
#define IMH    512
#define IMW    512

// atan2 for finite inputs, ~4 ulp, using v_rcp_f32 (no IEEE divide).
// Matches arctan2(y, x) conventions incl. signed zero; (0,0) -> 0.
__device__ __forceinline__ float fast_atan2f(float y, float x)
{
    const float ax = fabsf(x), ay = fabsf(y);
    const float mn = fminf(ax, ay), mx = fmaxf(ax, ay);
    // octant reduction: atan(t) = pi/4 + atan((t-1)/(t+1)), t = mn/mx,
    // with numerator/denominator selected BEFORE a single reciprocal.
    const bool  gt  = mn > 0.4142135623730950488f * mx;   // t > tan(pi/8)
    const float num = gt ? (mn - mx) : mn;
    const float den = gt ? (mn + mx) : mx;
    const float q   = num * __builtin_amdgcn_rcpf(den);   // v_rcp_f32, ~1 ulp
    const float z   = (den == 0.0f) ? 0.0f : q;           // guard 0 * inf
    const float s   = z * z;
    float p = 8.05374449538e-2f;
    p = p * s - 1.38776856032e-1f;
    p = p * s + 1.99777106478e-1f;
    p = p * s - 3.33329491539e-1f;
    float r = p * s * z + z;                              // atan(z), odd
    r += gt ? 0.78539816339744830962f : 0.0f;             // + pi/4
    r = (ay > ax) ? 1.57079632679489661923f - r : r;      // undo min/max swap
    r = (x < 0.0f) ? 3.14159265358979323846f - r : r;     // quadrant
    return copysignf(r, y);                               // hemisphere / ±0
}

// One block: 256 threads, one 64x64 tile of one image.
// Thread (ty,tx) = (tid>>4, tid&15) owns a 4x4 pixel patch (one pool block
// = 2x2 thread quad). Histogram accumulated in LDS via ds_add_f32 atomics.
__global__ __launch_bounds__(256)
void hog_fused_kernel(const float* __restrict__ x, float* __restrict__ out)
{
    __shared__ __align__(16) float tile[TROWS * TPITCH]; // 17952 B
    __shared__ float hist[NBINS * 64];                   // [bin*64 + poolblk]

    const int tid = threadIdx.x;
    const int bx  = blockIdx.x;   // 0..7 tile col
    const int by  = blockIdx.y;   // 0..7 tile row
    const int n   = blockIdx.z;   // 0..15 batch

    const int x0 = bx * TILE;
    const int y0 = by * TILE;

    // Zero the pooled histogram.
    for (int i = tid; i < NBINS * 64; i += 256) hist[i] = 0.0f;

    // Zero-fill the tile only where the halo can fall off the image.
    if (bx == 0 || by == 0 || bx == 7 || by == 7) {
        for (int i = tid; i < TROWS * TPITCH; i += 256) tile[i] = 0.0f;
    }
    __syncthreads();

    // ---- CDNA5 async global->LDS staging of the 66x66 halo patch ----
    const float* src = x + (size_t)n * (IMH * IMW);
    for (int i = tid; i < TROWS * TROWS; i += 256) {
        const int r  = i / TROWS;
        const int c  = i - r * TROWS;
        const int gr = y0 + r - 1;
        const int gc = x0 + c - 1;
        if ((unsigned)gr < (unsigned)IMH && (unsigned)gc < (unsigned)IMW) {
            const unsigned long long gaddr =
                (unsigned long long)(size_t)(src + (size_t)gr * IMW + gc);
            // low 32 bits of the flat shared address == LDS byte offset
            const unsigned lds_off =
                (unsigned)(size_t)(&tile[r * TPITCH + c]);
            asm volatile("global_load_async_to_lds_b32 %0, %1, off"
                         :: "v"(lds_off), "v"(gaddr)
                         : "memory");
        }
    }
    asm volatile("s_wait_asynccnt 0" ::: "memory");
    __syncthreads();

    // ---- per-thread 6x6 register window, loaded as 2 x b128 per row ----
    const int tx  = tid & 15;
    const int ty  = tid >> 4;
    const int cx  = tx * 4;                      // multiple of 4 -> 16B aligned
    const int cy  = ty * 4;
    const int blk = (ty >> 1) * 8 + (tx >> 1);   // pool block of this patch

    float p[6][6];
#pragma unroll
    for (int r = 0; r < 6; ++r) {
        const float4 a = *reinterpret_cast<const float4*>(&tile[(cy + r) * TPITCH + cx]);
        const float4 b = *reinterpret_cast<const float4*>(&tile[(cy + r) * TPITCH + cx + 4]);
        p[r][0] = a.x; p[r][1] = a.y; p[r][2] = a.z; p[r][3] = a.w;
        p[r][4] = b.x; p[r][5] = b.y;               // b.z/b.w never consumed
    }

    // ---- separable Sobel: vertical smooth / vertical diff, shared by row ----
    float v1[4][6];   // [1,2,1] vertical  -> gx = v1[i][j] - v1[i][j+2]
    float v2[4][6];   // [1,0,-1] vertical -> gy = v2[i][j] + 2*v2[i][j+1] + v2[i][j+2]
#pragma unroll
    for (int i = 0; i < 4; ++i)
#pragma unroll
        for (int c = 0; c < 6; ++c) {
            v1[i][c] = p[i][c] + 2.0f * p[i + 1][c] + p[i + 2][c];
            v2[i][c] = p[i][c] - p[i + 2][c];
        }

    const float binscale = (float)NBINS / 3.14159265358979323846f;

#pragma unroll
    for (int i = 0; i < 4; ++i) {
#pragma unroll
        for (int j = 0; j < 4; ++j) {
            // cross-correlation (no kernel flip), matching conv_general_dilated
            const float gxv = v1[i][j] - v1[i][j + 2];
            const float gyv = v2[i][j] + 2.0f * v2[i][j + 1] + v2[i][j + 2];

            const float mag = __builtin_amdgcn_sqrtf(gxv * gxv + gyv * gyv);
            const float ph  = fast_atan2f(gxv, gyv);     // arctan2(gx, gy)
            const float pi_ = ph * binscale;             // in [-10, 10]

            const float f  = floorf(pi_);
            int fl = (int)f;                             // [-10, 10]
            int ce = fl + ((pi_ != f) ? 1 : 0);          // ceil via floor+frac
            fl += (fl < 0) ? 10 : 0;  fl = (fl >= 10) ? fl - 10 : fl;
            ce += (ce < 0) ? 10 : 0;  ce = (ce >= 10) ? ce - 10 : ce;

            atomicAdd(&hist[(fl << 6) + blk], mag);          // ds_add_f32
            atomicAdd(&hist[(ce << 6) + blk], 1.0f - mag);   // ds_add_f32
        }
    }
    __syncthreads();

    // ---- write pooled output: 640 values, coalesced over pool blocks ----
    const float inv64 = 1.0f / 64.0f;
    for (int i = tid; i < NBINS * 64; i += 256) {
        const int bin = i >> 6;
        const int pb  = i & 63;
        const int yo  = by * 8 + (pb >> 3);
        const int xo  = bx * 8 + (pb & 7);
        out[(((size_t)n * NBINS + bin) * 64 + yo) * 64 + xo] = hist[i] * inv64;
    }
}

extern "C" void kernel_launch(void* const* d_in, const int* in_sizes, int n_in,
                              void* d_out, int out_size, void* d_ws, size_t ws_size,
                              hipStream_t stream)
{
    (void)in_sizes; (void)n_in; (void)out_size; (void)d_ws; (void)ws_size;
    const float* x = (const float*)d_in[0];
    float* out     = (float*)d_out;

    dim3 grid(8, 8, 16);      // 8x8 tiles of 64x64, 16 images
    dim3 block(256, 1, 1);    // 8 wave32 per block
    hog_fused_kernel<<<grid, block, 0, stream>>>(x, out);
}